// izhikevech_85856396247225
// MI455X (gfx1250) — compile-verified
//
#include <hip/hip_runtime.h>
#include <stdint.h>

// ---------------------------------------------------------------------------
// Izhikevich single step, f32 elementwise over N neurons.
// Memory-bound: 20 B/neuron -> ~335 MB -> ~14.4 us floor at 23.3 TB/s.
// Strategy: B128 NT loads/stores, async global->LDS double-buffered pipeline
// (ASYNCcnt), skip unused `mem` input, wave32-friendly 256-thread blocks.
// ---------------------------------------------------------------------------

typedef float f4 __attribute__((ext_vector_type(4)));

#define IZ_K      0.4471817006977834f
#define IZ_A      0.0032799410036917333f
#define IZ_B      24.478421990208606f
#define IZ_VMIN  -66.46563513097735f
#define IZ_D      50.0f
#define IZ_C      38.0f
#define IZ_VR    -77.40291336465064f
#define IZ_VT    -44.90054428048817f
#define IZ_VPEAK  15.489726771001997f

#define IZ_BLOCK 256
#define IZ_ITER  4   // 4 pipeline stages per block; 4096 floats/stream/block

__device__ __forceinline__ void izhi1(float v, float u, float cur,
                                      float hdC, float hA,
                                      float& vn, float& un) {
  const bool  spike = v >= IZ_VPEAK;
  const float dv    = IZ_K * (v - IZ_VR) * (v - IZ_VT) - u + cur;
  const float v_int = __builtin_fmaf(hdC, dv, v);          // v + h*dv/C
  const float du    = __builtin_fmaf(IZ_B, v_int - IZ_VR, -u);
  const float u_int = __builtin_fmaf(hA, du, u);           // u + h*A*du
  vn = spike ? IZ_VMIN : v_int;
  un = spike ? (u + IZ_D) : u_int;
}

__device__ __forceinline__ void izhi4(const f4 vv, const f4 uu, const f4 ii,
                                      float hdC, float hA, f4& vn, f4& un) {
#pragma unroll
  for (int c = 0; c < 4; ++c) {
    float a, b;
    izhi1(vv[c], uu[c], ii[c], hdC, hA, a, b);
    vn[c] = a;
    un[c] = b;
  }
}

// ---------------------------------------------------------------------------
// Fast path: uniform chunks, async global->LDS (double-buffered) when the
// gfx1250 async builtins exist; otherwise direct B128 NT loads. Requires
// n % (4*IZ_BLOCK*IZ_ITER) == 0 (enforced by host).
// ---------------------------------------------------------------------------
__global__ __launch_bounds__(IZ_BLOCK) void izhi_async(
    const f4* __restrict__ v, const f4* __restrict__ u, const f4* __restrict__ I,
    const float* __restrict__ hp, f4* __restrict__ vout, f4* __restrict__ uout) {
  const float h   = hp[0];
  const float hdC = h / IZ_C;
  const float hA  = h * IZ_A;
  const int   t    = threadIdx.x;
  const int   base = blockIdx.x * (IZ_BLOCK * IZ_ITER);

#if defined(__HIP_DEVICE_COMPILE__) && \
    __has_builtin(__builtin_amdgcn_global_load_async_to_lds_b128) && \
    __has_builtin(__builtin_amdgcn_s_wait_asynccnt)
  // Builtin signature (from clang diagnostic): param0 is
  //   'int __attribute__((vector_size(16))) __device__ *'  (v4i, addrspace(1))
  // so cast global src to AS1 v4i* and the LDS dst to AS3 v4i*.
  typedef int v4i __attribute__((vector_size(16)));
  typedef __attribute__((address_space(1))) v4i* as1_v4i_p;
  typedef __attribute__((address_space(3))) v4i* as3_v4i_p;

  __shared__ f4 s_v[2][IZ_BLOCK];
  __shared__ f4 s_u[2][IZ_BLOCK];
  __shared__ f4 s_I[2][IZ_BLOCK];

#define IZ_ASYNC_LD(gp, lp) \
  __builtin_amdgcn_global_load_async_to_lds_b128((as1_v4i_p)(gp), (as3_v4i_p)(lp), 0, 0)

  // Prologue: stage 0 into buffer 0 (3 async instrs -> ASYNCcnt = 3).
  IZ_ASYNC_LD(v + base + t, &s_v[0][t]);
  IZ_ASYNC_LD(u + base + t, &s_u[0][t]);
  IZ_ASYNC_LD(I + base + t, &s_I[0][t]);

#pragma unroll
  for (int j = 0; j < IZ_ITER; ++j) {
    const int cur = j & 1;
    const int gi  = base + j * IZ_BLOCK + t;
    if (j + 1 < IZ_ITER) {
      const int gn = gi + IZ_BLOCK;
      IZ_ASYNC_LD(v + gn, &s_v[cur ^ 1][t]);
      IZ_ASYNC_LD(u + gn, &s_u[cur ^ 1][t]);
      IZ_ASYNC_LD(I + gn, &s_I[cur ^ 1][t]);
      __builtin_amdgcn_s_wait_asynccnt(3);   // stage j landed; j+1 in flight
    } else {
      __builtin_amdgcn_s_wait_asynccnt(0);
    }
    // Each lane reads only its own 16B LDS slot -> no workgroup barrier needed.
    const f4 vv = s_v[cur][t];
    const f4 uu = s_u[cur][t];
    const f4 ii = s_I[cur][t];
    f4 vn, un;
    izhi4(vv, uu, ii, hdC, hA, vn, un);
    __builtin_nontemporal_store(vn, vout + gi);
    __builtin_nontemporal_store(un, uout + gi);
  }
#undef IZ_ASYNC_LD
#else
  // Fallback body (also what the host-side pass sees): direct B128 NT loads.
#pragma unroll
  for (int j = 0; j < IZ_ITER; ++j) {
    const int gi = base + j * IZ_BLOCK + t;
    const f4 vv = __builtin_nontemporal_load(v + gi);
    const f4 uu = __builtin_nontemporal_load(u + gi);
    const f4 ii = __builtin_nontemporal_load(I + gi);
    f4 vn, un;
    izhi4(vv, uu, ii, hdC, hA, vn, un);
    __builtin_nontemporal_store(vn, vout + gi);
    __builtin_nontemporal_store(un, uout + gi);
  }
#endif
}

// ---------------------------------------------------------------------------
// Generic vector path: grid-stride over f4 elements (n % 4 == 0).
// ---------------------------------------------------------------------------
__global__ __launch_bounds__(IZ_BLOCK) void izhi_vec(
    const f4* __restrict__ v, const f4* __restrict__ u, const f4* __restrict__ I,
    const float* __restrict__ hp, f4* __restrict__ vout, f4* __restrict__ uout,
    int n4) {
  const float h   = hp[0];
  const float hdC = h / IZ_C;
  const float hA  = h * IZ_A;
  const int stride = gridDim.x * blockDim.x;
  for (int i = blockIdx.x * blockDim.x + threadIdx.x; i < n4; i += stride) {
    __builtin_prefetch(v + i + stride, 0, 0);  // speculative global_prefetch_b8
    const f4 vv = __builtin_nontemporal_load(v + i);
    const f4 uu = __builtin_nontemporal_load(u + i);
    const f4 ii = __builtin_nontemporal_load(I + i);
    f4 vn, un;
    izhi4(vv, uu, ii, hdC, hA, vn, un);
    __builtin_nontemporal_store(vn, vout + i);
    __builtin_nontemporal_store(un, uout + i);
  }
}

// ---------------------------------------------------------------------------
// Scalar fallback for arbitrary n.
// ---------------------------------------------------------------------------
__global__ __launch_bounds__(IZ_BLOCK) void izhi_scalar(
    const float* __restrict__ v, const float* __restrict__ u,
    const float* __restrict__ I, const float* __restrict__ hp,
    float* __restrict__ vout, float* __restrict__ uout, int n) {
  const float h   = hp[0];
  const float hdC = h / IZ_C;
  const float hA  = h * IZ_A;
  const int i = blockIdx.x * blockDim.x + threadIdx.x;
  if (i < n) {
    float a, b;
    izhi1(v[i], u[i], I[i], hdC, hA, a, b);
    vout[i] = a;
    uout[i] = b;
  }
}

extern "C" void kernel_launch(void* const* d_in, const int* in_sizes, int n_in,
                              void* d_out, int out_size, void* d_ws, size_t ws_size,
                              hipStream_t stream) {
  (void)n_in; (void)out_size; (void)d_ws; (void)ws_size;
  const float* v = (const float*)d_in[0];
  const float* u = (const float*)d_in[1];
  const float* I = (const float*)d_in[2];
  // d_in[3] = mem: unused by the reference -> never read (saves 64 MB traffic)
  const float* h = (const float*)d_in[4];

  const int n = in_sizes[0];
  float* vout = (float*)d_out;
  float* uout = vout + n;

  const int chunk = 4 * IZ_BLOCK * IZ_ITER;  // floats per block on the fast path
  if (n > 0 && (n % chunk) == 0) {
    const int blocks = n / chunk;            // N=16.7M -> 4096 blocks x 8 waves
    izhi_async<<<blocks, IZ_BLOCK, 0, stream>>>(
        (const f4*)v, (const f4*)u, (const f4*)I, h, (f4*)vout, (f4*)uout);
  } else if (n > 0 && (n % 4) == 0) {
    const int n4 = n / 4;
    int blocks = (n4 + IZ_BLOCK - 1) / IZ_BLOCK;
    if (blocks > 16384) blocks = 16384;
    izhi_vec<<<blocks, IZ_BLOCK, 0, stream>>>(
        (const f4*)v, (const f4*)u, (const f4*)I, h, (f4*)vout, (f4*)uout, n4);
  } else if (n > 0) {
    const int blocks = (n + IZ_BLOCK - 1) / IZ_BLOCK;
    izhi_scalar<<<blocks, IZ_BLOCK, 0, stream>>>(v, u, I, h, vout, uout, n);
  }
}